// GCN_90134183673917
// MI455X (gfx1250) — compile-verified
//
#include <hip/hip_runtime.h>
#include <hip/hip_bf16.h>

typedef __attribute__((ext_vector_type(2))) float v2f;
typedef __attribute__((ext_vector_type(8))) float v8f;

#define NUM_GRAPHS 512
#define NODES_PER_GRAPH 200
#define N_NODES (NUM_GRAPHS * NODES_PER_GRAPH)   // 102400
#define EDGES_PER_GRAPH 3200
#define N_EDGES (NUM_GRAPHS * EDGES_PER_GRAPH)   // 1638400
#define KDIM 200                                 // N_rois (input feature dim)
#define HID 64
#define OUTD 2
#define TPB 256

#if defined(__has_builtin)
#if __has_builtin(__builtin_amdgcn_global_load_async_to_lds_b128)
#define HAVE_ASYNC_LDS 1
#endif
#endif

#ifdef HAVE_ASYNC_LDS
// builtin signature: (global int4*, lds int4*, imm offset, imm cpol)
typedef int async_v4i __attribute__((__vector_size__(16)));
typedef __attribute__((address_space(1))) async_v4i* g_async_p;
typedef __attribute__((address_space(3))) async_v4i* l_async_p;
#endif

// ---------------------------------------------------------------------------
// GEMM: h[N,64] = x[N,200] @ conv_w[200,64] via V_WMMA_F32_16X16X4_F32.
// 4 waves per block; each wave computes a 16-row x 64-col tile.
// conv_w staged in LDS (200*64*4 = 50 KB; WGP LDS = 320 KB).
// ---------------------------------------------------------------------------
__global__ void __launch_bounds__(128) gemm_wmma_kernel(const float* __restrict__ x,
                                                        const float* __restrict__ w,
                                                        float* __restrict__ h) {
    __shared__ float lw[KDIM * HID];  // 50 KB
    int t = threadIdx.x;
    for (int i = t; i < KDIM * HID; i += 128) lw[i] = w[i];
    __syncthreads();

    const int wave = t >> 5;
    const int lane = t & 31;
    const int m0   = (blockIdx.x * 4 + wave) * 16;     // row tile base
    const int mrow = m0 + (lane & 15);
    const int ksel = (lane >> 4) * 2;                  // lanes 0-15: K {0,1}; 16-31: K {2,3}
    const float* xrow = x + (size_t)mrow * KDIM;

#pragma unroll
    for (int j = 0; j < 4; ++j) {                      // 4 N-tiles of 16 cover HID=64
        const int n0  = j * 16;
        const int col = n0 + (lane & 15);
        v8f c = {};
        for (int k = 0; k < KDIM; k += 4) {            // 50 K-steps
            v2f a, b;
            a.x = xrow[k + ksel];
            a.y = xrow[k + ksel + 1];
            b.x = lw[(k + ksel) * HID + col];
            b.y = lw[(k + ksel + 1) * HID + col];
            c = __builtin_amdgcn_wmma_f32_16x16x4_f32(
                    /*neg_a=*/false, a, /*neg_b=*/false, b,
                    /*c_mod=*/(short)0, c, /*reuse_a=*/false, /*reuse_b=*/false);
        }
        const int base_m = m0 + ((lane >> 4) * 8);
#pragma unroll
        for (int v = 0; v < 8; ++v) {
            h[(size_t)(base_m + v) * HID + col] = c[v];
        }
    }
}

// ---------------------------------------------------------------------------
// Fused per-graph pipeline: one 256-thread workgroup per graph.
// LDS holds h (51.2 KB, async-staged) + agg (51.2 KB) + dinv + reduction
// buffers (~104 KB total << 320 KB WGP LDS).  All scatter-adds are LDS
// atomics (ds_add_f32); h is read from HBM exactly once, coalesced.
// ---------------------------------------------------------------------------
__global__ void __launch_bounds__(TPB) fused_graph_kernel(const int* __restrict__ esrc_all,
                                                          const int* __restrict__ edst_all,
                                                          const float* __restrict__ h,
                                                          const float* __restrict__ conv_b,
                                                          const float* __restrict__ lin_w,
                                                          const float* __restrict__ lin_b,
                                                          float* __restrict__ out_head,
                                                          float* __restrict__ out_pool) {
    __shared__ __align__(16) float h_lds[NODES_PER_GRAPH * HID];  // 51.2 KB
    __shared__ __align__(16) float agg[NODES_PER_GRAPH * HID];    // 51.2 KB
    __shared__ float dinv[NODES_PER_GRAPH];                       // deg -> dinv in place
    __shared__ float pmax[4 * HID];
    __shared__ float pool[HID];

    const int g     = blockIdx.x;
    const int t     = threadIdx.x;
    const int node0 = g * NODES_PER_GRAPH;
    const int* esrc = esrc_all + (size_t)g * EDGES_PER_GRAPH;
    const int* edst = edst_all + (size_t)g * EDGES_PER_GRAPH;

    // ---- stage h block into LDS (coalesced b128 per lane), async if available
    {
        const float4* hg  = (const float4*)(h + (size_t)node0 * HID);
        float4*       hl4 = (float4*)h_lds;
        for (int i = t; i < NODES_PER_GRAPH * HID / 4; i += TPB) {
#ifdef HAVE_ASYNC_LDS
            __builtin_amdgcn_global_load_async_to_lds_b128(
                (g_async_p)(hg + i), (l_async_p)(hl4 + i), 0, 0);
#else
            hl4[i] = hg[i];
#endif
        }
    }

    // ---- degree count in LDS (overlaps the async h staging)
    for (int i = t; i < NODES_PER_GRAPH; i += TPB) dinv[i] = 1.0f;  // self loop
    __syncthreads();
    for (int e = t; e < EDGES_PER_GRAPH; e += TPB) {
        atomicAdd(&dinv[edst[e] - node0], 1.0f);                    // ds_add_f32
    }

#ifdef HAVE_ASYNC_LDS
#if __has_builtin(__builtin_amdgcn_s_wait_asynccnt)
    __builtin_amdgcn_s_wait_asynccnt(0);
#else
    asm volatile("s_wait_asynccnt 0x0" ::: "memory");
#endif
#endif
    __syncthreads();

    for (int i = t; i < NODES_PER_GRAPH; i += TPB) dinv[i] = rsqrtf(dinv[i]);
    __syncthreads();

    // ---- agg = self-loop term + bias
    for (int i = t; i < NODES_PER_GRAPH * HID; i += TPB) {
        int n   = i >> 6;
        int f   = i & 63;
        float d = dinv[n];
        agg[i] = h_lds[i] * d * d + conv_b[f];
    }
    __syncthreads();

    // ---- edge scatter entirely in LDS: 3200 edges x 16 float4-quads
    for (int w = t; w < EDGES_PER_GRAPH * (HID / 4); w += TPB) {
        int e  = w >> 4;
        int f4 = (w & 15) * 4;
        int s  = esrc[e] - node0;
        int d  = edst[e] - node0;
        float nrm = dinv[s] * dinv[d];
        float4 v  = *(const float4*)&h_lds[s * HID + f4];
        float* ad = &agg[d * HID + f4];
        atomicAdd(ad + 0, v.x * nrm);
        atomicAdd(ad + 1, v.y * nrm);
        atomicAdd(ad + 2, v.z * nrm);
        atomicAdd(ad + 3, v.w * nrm);
    }
    __syncthreads();

    // ---- relu + segment-max pool (relu folds into max since mx starts at 0)
    {
        int f = t & 63;
        int c = t >> 6;  // 4 chunks of 50 nodes
        float mx = 0.0f;
        for (int n = c * 50; n < (c + 1) * 50; ++n) {
            mx = fmaxf(mx, agg[n * HID + f]);
        }
        pmax[c * HID + f] = mx;
    }
    __syncthreads();
    if (t < HID) {
        float mx = fmaxf(fmaxf(pmax[t], pmax[HID + t]),
                         fmaxf(pmax[2 * HID + t], pmax[3 * HID + t]));
        out_pool[(size_t)g * HID + t] = mx;
        pool[t] = mx;
    }
    __syncthreads();

    // ---- 64x2 linear head
    if (t < OUTD) {
        float s = lin_b[t];
#pragma unroll
        for (int i = 0; i < HID; ++i) s += pool[i] * lin_w[i * OUTD + t];
        out_head[g * OUTD + t] = s;
    }
}

// ---------------------------------------------------------------------------
extern "C" void kernel_launch(void* const* d_in, const int* in_sizes, int n_in,
                              void* d_out, int out_size, void* d_ws, size_t ws_size,
                              hipStream_t stream) {
    const float* x      = (const float*)d_in[0];
    const int*   eidx   = (const int*)d_in[1];   // [2, E] int32
    // d_in[2] = batch (unused: nodes are block-contiguous per graph)
    const float* conv_w = (const float*)d_in[3];
    const float* conv_b = (const float*)d_in[4];
    const float* lin_w  = (const float*)d_in[5];
    const float* lin_b  = (const float*)d_in[6];

    const int* src = eidx;            // row 0
    const int* dst = eidx + N_EDGES;  // row 1

    float* h = (float*)d_ws;          // N*64 floats (26.2 MB)

    float* out_head = (float*)d_out;                  // [512, 2]
    float* out_pool = out_head + NUM_GRAPHS * OUTD;   // [512, 64]

    // h = x @ conv_w  (WMMA f32): 102400/16 = 6400 row-tiles, 4 waves/block
    gemm_wmma_kernel<<<N_NODES / 16 / 4, 128, 0, stream>>>(x, conv_w, h);

    // fused per-graph: degree -> dinv -> LDS aggregate -> pool -> head
    fused_graph_kernel<<<NUM_GRAPHS, TPB, 0, stream>>>(src, dst, h, conv_b, lin_w, lin_b,
                                                       out_head, out_pool);
}